// FuzzyMergeLayer_10857677324318
// MI455X (gfx1250) — compile-verified
//
#include <hip/hip_runtime.h>
#include <hip/hip_bf16.h>

typedef float    v2f __attribute__((ext_vector_type(2)));
typedef float    v4f __attribute__((ext_vector_type(4)));
typedef float    v8f __attribute__((ext_vector_type(8)));
typedef unsigned v4u __attribute__((ext_vector_type(4)));
typedef int      v4i __attribute__((ext_vector_type(4)));
typedef int      v8i __attribute__((ext_vector_type(8)));

#define NPIX      (8 * 256 * 256)   // 524288 pixels
#define CCH       16
#define KCLS      8
#define SUPER     64                // pixels per TDM tile
#define NSUPER    (NPIX / SUPER)    // 8192
#define ROWSTRIDE 18                // dwords per LDS row (16 data + 2 pad from TDM)
#define WAVES     4
#define BLOCKT    (WAVES * 32)
#define GRID      256

#define MEMBAR() asm volatile("" ::: "memory")

// ---------------------------------------------------------------------------
// Kernel 0: per-class precompute (triangular inverse, bias, log-det constant)
// W[k][i][c]  = Linv_k[i][c]           (A-matrix for WMMA)
// bneg[k][i]  = -(Linv_k @ mu_k)[i]    (C-init bias, z = Linv x + bneg)
// cK[k]       = -8*log(2*pi) - sum_i log|L[i][i]|
// ---------------------------------------------------------------------------
__global__ __launch_bounds__(128) void mvn_precompute(
    const float* __restrict__ mean, const float* __restrict__ tril,
    float* __restrict__ W, float* __restrict__ bneg, float* __restrict__ cK) {
  const int t = threadIdx.x;          // 0..127
  const int k = t >> 4, j = t & 15;   // class, column
  const float* L = tril + k * 256;

  // forward substitution: column j of inv(tril(L_k))
  float z[16];
  for (int i = 0; i < 16; ++i) {
    if (i < j) { z[i] = 0.0f; continue; }
    float s = (i == j) ? 1.0f : 0.0f;
    for (int m = j; m < i; ++m) s -= L[i * 16 + m] * z[m];
    z[i] = s / L[i * 16 + i];
  }
  for (int i = 0; i < 16; ++i) W[(k * 16 + i) * 16 + j] = z[i];
  __syncthreads();

  // bias row j: bneg[k][j] = -sum_c Linv[k][j][c] * mean[k][c]
  float s = 0.0f;
  for (int c = 0; c < 16; ++c) s += W[(k * 16 + j) * 16 + c] * mean[k * 16 + c];
  bneg[k * 16 + j] = -s;

  if (j == 0) {
    float ld = 0.0f;
    for (int i = 0; i < 16; ++i) ld += __logf(fabsf(L[i * 16 + i]));
    cK[k] = -8.0f * 1.8378770664093453f - ld;   // -0.5*C*log(2pi) - logdet
  }
}

// ---------------------------------------------------------------------------
// TDM issue: load SUPER x 16 fp32 tile (row-major, 16 dwords/row) into LDS
// with 2-dword padding per row -> 18-dword LDS row stride.
// ---------------------------------------------------------------------------
__device__ __forceinline__ void tdm_issue(const float* xbase, int task,
                                          unsigned ldsoff) {
  unsigned long long ga = (unsigned long long)(uintptr_t)xbase +
                          (unsigned long long)task * (SUPER * CCH * 4);
  v4u g0;
  g0.x = 1u;                                            // count = 1 (valid)
  g0.y = ldsoff;                                        // LDS byte address
  g0.z = (unsigned)ga;                                  // global_addr[31:0]
  g0.w = (unsigned)((ga >> 32) & 0x01FFFFFFu) | 0x80000000u;  // addr[56:32] | type=2

  v8i g1;
  // data_size=4B(2), pad_enable, pad_interval=3 (16 dwords), pad_amount=1 (2 dwords)
  g1[0] = (2 << 16) | (1 << 20) | (3 << 22) | (1 << 25);
  g1[1] = (CCH << 16);                                  // tensor_dim0 = 16
  g1[2] = 0;                                            // dim0_hi=0, dim1_lo16=0
  g1[3] = (NPIX >> 16) | (CCH << 16);                   // tensor_dim1_hi, tile_dim0=16
  g1[4] = SUPER;                                        // tile_dim1 = 64, tile_dim2 = 0
  g1[5] = CCH;                                          // tensor_dim0_stride = 16
  g1[6] = 0;
  g1[7] = 0;

  v4i z4 = {0, 0, 0, 0};
  v8i z8 = {0, 0, 0, 0, 0, 0, 0, 0};
  __builtin_amdgcn_tensor_load_to_lds(g0, g1, z4, z4, z8, 0);
}

// ---------------------------------------------------------------------------
// Main kernel: WMMA whitening + exp + L2-normalize + fused concat store.
// Each wave owns two private LDS buffers (double-buffered TDM pipeline).
// ---------------------------------------------------------------------------
__global__ __launch_bounds__(BLOCKT) void mvn_main(
    const float* __restrict__ x, const float* __restrict__ W,
    const float* __restrict__ bneg, const float* __restrict__ cKp,
    float* __restrict__ out, int nwavesTotal) {
  __shared__ float sh[WAVES][2][SUPER * ROWSTRIDE];

  const int lane = threadIdx.x & 31;
  const int wave = threadIdx.x >> 5;
  const int wid  = blockIdx.x * WAVES + wave;
  const int p    = lane & 15;     // pixel-in-subtile / matrix row
  const int half = lane >> 4;     // K-split half

  // Resident A fragments: A[k][f] = {Linv_k[p][4f+2h], Linv_k[p][4f+2h+1]}
  v2f A[KCLS][4];
#pragma unroll
  for (int k = 0; k < KCLS; ++k)
#pragma unroll
    for (int f = 0; f < 4; ++f)
      A[k][f] = *(const v2f*)(W + (k * 16 + p) * 16 + 4 * f + 2 * half);

  // Resident C-init bias vectors: bias[k][v] = -b_k[v + 8*half]
  v8f bias[KCLS];
#pragma unroll
  for (int k = 0; k < KCLS; ++k) {
    v4f lo = *(const v4f*)(bneg + k * 16 + 8 * half);
    v4f hi = *(const v4f*)(bneg + k * 16 + 8 * half + 4);
    v8f b;
    b[0] = lo[0]; b[1] = lo[1]; b[2] = lo[2]; b[3] = lo[3];
    b[4] = hi[0]; b[5] = hi[1]; b[6] = hi[2]; b[7] = hi[3];
    bias[k] = b;
  }
  float cK[KCLS];
#pragma unroll
  for (int k = 0; k < KCLS; ++k) cK[k] = cKp[k];

  const unsigned lds0 = (unsigned)(uintptr_t)&sh[wave][0][0];
  const unsigned lds1 = (unsigned)(uintptr_t)&sh[wave][1][0];

  if (wid < NSUPER) tdm_issue(x, wid, lds0);
  MEMBAR();

  int buf = 0;
  for (int task = wid; task < NSUPER; task += nwavesTotal) {
    const int nxt = task + nwavesTotal;
    if (nxt < NSUPER) {
      tdm_issue(x, nxt, buf ? lds0 : lds1);
      MEMBAR();
      __builtin_amdgcn_s_wait_tensorcnt(1);   // current buffer landed
    } else {
      __builtin_amdgcn_s_wait_tensorcnt(0);
    }
    MEMBAR();

    const float* xs = buf ? &sh[wave][1][0] : &sh[wave][0][0];

    for (int sub = 0; sub < SUPER / 16; ++sub) {
      const float* row = xs + (sub * 16 + p) * ROWSTRIDE;

      // B fragments: Bf[f] = {x[p][4f+2h], x[p][4f+2h+1]} (8B-aligned ds_load)
      v2f Bf[4];
#pragma unroll
      for (int f = 0; f < 4; ++f)
        Bf[f] = *(const v2f*)(row + 4 * f + 2 * half);

      float pr[KCLS];
      float ssum = 0.0f;
#pragma unroll
      for (int k = 0; k < KCLS; ++k) {
        v8f acc = bias[k];                    // C-init with -b_k  ->  D = z
#pragma unroll
        for (int f = 0; f < 4; ++f)
          acc = __builtin_amdgcn_wmma_f32_16x16x4_f32(
              false, A[k][f], false, Bf[f], (short)0, acc, false, false);

        float s2 = 0.0f;
#pragma unroll
        for (int i = 0; i < 8; ++i) s2 += acc[i] * acc[i];
        // merge row-halves: lanes l <-> l+16 (SWAPX16 swizzle)
        float oth = __int_as_float(
            __builtin_amdgcn_ds_swizzle(__float_as_int(s2), 0x401F));
        float quad = s2 + oth;
        float pk = __expf(cK[k] - 0.5f * quad);
        pr[k] = pk;
        ssum += pk * pk;
      }

      const float rn = __frsqrt_rn(fmaxf(ssum, 1e-12f));
#pragma unroll
      for (int k = 0; k < KCLS; ++k) pr[k] *= rn;

      const int n = task * SUPER + sub * 16 + p;
      float* orow = out + (long long)n * (CCH + KCLS);

      // x passthrough: each half copies 8 channels (2x b128, 16B-aligned)
      v2f x0 = *(const v2f*)(row + 8 * half + 0);
      v2f x1 = *(const v2f*)(row + 8 * half + 2);
      v2f x2 = *(const v2f*)(row + 8 * half + 4);
      v2f x3 = *(const v2f*)(row + 8 * half + 6);
      v4f xa; xa[0] = x0[0]; xa[1] = x0[1]; xa[2] = x1[0]; xa[3] = x1[1];
      v4f xb; xb[0] = x2[0]; xb[1] = x2[1]; xb[2] = x3[0]; xb[3] = x3[1];
      *(v4f*)(orow + 8 * half)     = xa;
      *(v4f*)(orow + 8 * half + 4) = xb;

      if (half == 0) {   // prob values identical in both halves; one writes
        v4f pa; pa[0] = pr[0]; pa[1] = pr[1]; pa[2] = pr[2]; pa[3] = pr[3];
        v4f pb; pb[0] = pr[4]; pb[1] = pr[5]; pb[2] = pr[6]; pb[3] = pr[7];
        *(v4f*)(orow + 16) = pa;
        *(v4f*)(orow + 20) = pb;
      }
    }
    buf ^= 1;
  }
}

// ---------------------------------------------------------------------------
extern "C" void kernel_launch(void* const* d_in, const int* in_sizes, int n_in,
                              void* d_out, int out_size, void* d_ws,
                              size_t ws_size, hipStream_t stream) {
  const float* x    = (const float*)d_in[0];   // [8,256,256,16]
  const float* mean = (const float*)d_in[1];   // [8,16]
  const float* tril = (const float*)d_in[2];   // [8,16,16]
  float* out = (float*)d_out;                  // [8,256,256,24]

  float* W    = (float*)d_ws;                  // 2048 floats
  float* bneg = W + KCLS * CCH * CCH;          // 128 floats
  float* cK   = bneg + KCLS * CCH;             // 8 floats

  mvn_precompute<<<1, 128, 0, stream>>>(mean, tril, W, bneg, cK);

  const int nwaves = GRID * WAVES;
  mvn_main<<<GRID, BLOCKT, 0, stream>>>(x, W, bneg, cK, out, nwaves);
}